// ResBlock_81896436400577
// MI455X (gfx1250) — compile-verified
//
#include <hip/hip_runtime.h>
#include <cstdint>
#include <cstddef>

typedef __attribute__((ext_vector_type(2))) float v2f;
typedef __attribute__((ext_vector_type(8))) float v8f;

namespace {
constexpr int BATCH = 128;
constexpr int NEDGE = 50000;
constexpr int NNODE = 2000;
constexpr int KCH   = 16;
constexpr int CCH   = NNODE * KCH;   // 32000
constexpr float EPSV = 1e-5f;
}

// ---------------- utility kernels ----------------

__global__ void k_zero_i32(int* __restrict__ p, int n) {
    int i = blockIdx.x * 256 + threadIdx.x;
    if (i < n) p[i] = 0;
}

// edge_dst[e] = w1_dst[e*16]/16 ; edge_src[e] = w3_src[e*16]/16 ; histogram deg by dst
__global__ void k_prep(const int* __restrict__ w1_dst, const int* __restrict__ w3_src,
                       int* __restrict__ edge_dst, int* __restrict__ edge_src,
                       int* __restrict__ deg) {
    int e = blockIdx.x * 256 + threadIdx.x;
    if (e >= NEDGE) return;
    int d = w1_dst[(size_t)e * KCH] >> 4;   // /K, K=16
    int s = w3_src[(size_t)e * KCH] >> 4;
    edge_dst[e] = d;
    edge_src[e] = s;
    atomicAdd(&deg[d], 1);
}

// exclusive scan of deg[NNODE] -> rowptr[NNODE+1], one wave32
__global__ void k_scan(const int* __restrict__ deg, int* __restrict__ rowptr) {
    const int lane = threadIdx.x;                 // 0..31
    const int chunk = (NNODE + 31) / 32;
    const int s0 = lane * chunk;
    const int s1 = (s0 + chunk < NNODE) ? (s0 + chunk) : NNODE;
    int s = 0;
    for (int i = s0; i < s1; ++i) s += deg[i];
    int incl = s;
    for (int d = 1; d < 32; d <<= 1) {
        int t = __shfl_up(incl, d, 32);
        if (lane >= d) incl += t;
    }
    int run = incl - s;                           // exclusive prefix of this chunk
    for (int i = s0; i < s1; ++i) { rowptr[i] = run; run += deg[i]; }
    if (lane == 31) rowptr[NNODE] = run;          // == NEDGE
}

__global__ void k_fill(const int* __restrict__ edge_dst, const int* __restrict__ rowptr,
                       int* __restrict__ cnt, int* __restrict__ csr) {
    int e = blockIdx.x * 256 + threadIdx.x;
    if (e >= NEDGE) return;
    int n = edge_dst[e];
    int pos = rowptr[n] + atomicAdd(&cnt[n], 1);
    csr[pos] = e;
}

// deterministic order: sort each node's edge list ascending (avg deg ~25)
__global__ void k_sort(const int* __restrict__ rowptr, int* __restrict__ csr) {
    int n = blockIdx.x * 256 + threadIdx.x;
    if (n >= NNODE) return;
    int beg = rowptr[n], end = rowptr[n + 1];
    for (int i = beg + 1; i < end; ++i) {
        int v = csr[i];
        int j = i - 1;
        while (j >= beg && csr[j] > v) { csr[j + 1] = csr[j]; --j; }
        csr[j + 1] = v;
    }
}

// x[B][E] -> xT[E][B], 32x32 LDS tiles
__global__ void k_transpose(const float* __restrict__ x, float* __restrict__ xT) {
    __shared__ float tile[32][33];
    const int e0 = blockIdx.x * 32, b0 = blockIdx.y * 32;
    const int tx = threadIdx.x, ty = threadIdx.y;   // 32 x 8
    #pragma unroll
    for (int j = 0; j < 32; j += 8) {
        int e = e0 + tx, b = b0 + ty + j;
        tile[ty + j][tx] = (e < NEDGE) ? x[(size_t)b * NEDGE + e] : 0.0f;
    }
    __syncthreads();
    #pragma unroll
    for (int j = 0; j < 32; j += 8) {
        int e = e0 + ty + j, b = b0 + tx;
        if (e < NEDGE) xT[(size_t)e * BATCH + b] = tile[tx][ty + j];
    }
}

// ---------------- fused lin1 + group-LN + ELU (WMMA f32 16x16x4) ----------------
// One wave per (node n, batch tile of 16). D(16b x 16k) = sum over edge chunks of
// A(16x4) x B(4x16), A[m][kk] = x[b0+m, e_kk], B[kk][k] = w1[e_kk*16 + k].
// A/B VGPR layout: VGPR0 holds K=0 (lanes 0-15) / K=2 (lanes 16-31), VGPR1 K=1/3.

template <bool USE_XT>
__device__ __forceinline__ v8f lin1_chunk(v8f c, const float* __restrict__ xsrc,
                                          const float* __restrict__ w1,
                                          const int* __restrict__ csr,
                                          int p, int half, int bm, int m) {
    const int e0 = csr[p + half * 2];
    const int e1 = csr[p + half * 2 + 1];
    v2f av, bv;
    av.x = USE_XT ? xsrc[(size_t)e0 * BATCH + bm] : xsrc[(size_t)bm * NEDGE + e0];
    av.y = USE_XT ? xsrc[(size_t)e1 * BATCH + bm] : xsrc[(size_t)bm * NEDGE + e1];
    bv.x = w1[(size_t)e0 * KCH + m];
    bv.y = w1[(size_t)e1 * KCH + m];
    return __builtin_amdgcn_wmma_f32_16x16x4_f32(false, av, false, bv,
                                                 (short)0, c, false, false);
}

// tail chunk: clamp indices into valid range, zero the B rows of padded K-slices
template <bool USE_XT>
__device__ __forceinline__ v8f lin1_chunk_tail(v8f c, const float* __restrict__ xsrc,
                                               const float* __restrict__ w1,
                                               const int* __restrict__ csr,
                                               int p, int beg, int end,
                                               int half, int bm, int m) {
    const int i0 = p + half * 2;
    const int i1 = i0 + 1;
    const float m0 = (i0 < end) ? 1.0f : 0.0f;
    const float m1 = (i1 < end) ? 1.0f : 0.0f;
    const int e0 = csr[(i0 < end) ? i0 : beg];
    const int e1 = csr[(i1 < end) ? i1 : beg];
    v2f av, bv;
    av.x = USE_XT ? xsrc[(size_t)e0 * BATCH + bm] : xsrc[(size_t)bm * NEDGE + e0];
    av.y = USE_XT ? xsrc[(size_t)e1 * BATCH + bm] : xsrc[(size_t)bm * NEDGE + e1];
    bv.x = w1[(size_t)e0 * KCH + m] * m0;
    bv.y = w1[(size_t)e1 * KCH + m] * m1;
    return __builtin_amdgcn_wmma_f32_16x16x4_f32(false, av, false, bv,
                                                 (short)0, c, false, false);
}

template <bool USE_XT>
__global__ __launch_bounds__(32) void k_lin1_norm(
        const float* __restrict__ xsrc, const float* __restrict__ w1,
        const float* __restrict__ b1, const float* __restrict__ gamma,
        const float* __restrict__ beta, const int* __restrict__ rowptr,
        const int* __restrict__ csr, float* __restrict__ h) {
    const int n    = blockIdx.x;
    const int b0   = blockIdx.y * 16;
    const int lane = threadIdx.x;
    const int half = lane >> 4;
    const int m    = lane & 15;
    const int bm   = b0 + m;
    const int beg = rowptr[n], end = rowptr[n + 1];
    const int deg = end - beg;

    v8f c0 = {};
    v8f c1 = {};
    int p = beg;
    // two independent accumulators: chunk k+1's loads overlap chunk k's WMMA
    const int nDouble = deg >> 3;
    for (int it = 0; it < nDouble; ++it, p += 8) {
        c0 = lin1_chunk<USE_XT>(c0, xsrc, w1, csr, p,     half, bm, m);
        c1 = lin1_chunk<USE_XT>(c1, xsrc, w1, csr, p + 4, half, bm, m);
    }
    if (deg & 4) {                       // wave-uniform branch
        c0 = lin1_chunk<USE_XT>(c0, xsrc, w1, csr, p, half, bm, m);
        p += 4;
    }
    if (deg & 3) {                       // wave-uniform branch
        c1 = lin1_chunk_tail<USE_XT>(c1, xsrc, w1, csr, p, beg, end, half, bm, m);
    }

    const float g    = gamma[n];
    const float bb   = beta[n];
    const float bias = b1[n * KCH + m];
    #pragma unroll
    for (int r = 0; r < 8; ++r) {
        // C/D layout: VGPR r, lanes 0-15 -> row M=r, lanes 16-31 -> row M=8+r; col N = lane%16
        float v = c0[r] + c1[r] + bias;
        float s1 = v, s2 = v * v;
        #pragma unroll
        for (int mask = 8; mask >= 1; mask >>= 1) {   // reduce over the 16 channels
            s1 += __shfl_xor(s1, mask, 16);
            s2 += __shfl_xor(s2, mask, 16);
        }
        const float mean = s1 * (1.0f / 16.0f);
        float var = s2 * (1.0f / 16.0f) - mean * mean;
        var = var < 0.0f ? 0.0f : var;
        const float inv = rsqrtf(var + EPSV);
        float o = (v - mean) * inv * g + bb;
        o = (o > 0.0f) ? o : expm1f(o);               // ELU (alpha=1)
        h[(size_t)(b0 + half * 8 + r) * CCH + n * KCH + m] = o;
    }
}

// ---------------- lin3: per-edge gather dot + bias + residual ----------------
__global__ __launch_bounds__(256) void k_lin3(
        const float* __restrict__ x, const float* __restrict__ h,
        const float* __restrict__ w3, const float* __restrict__ b3,
        const int* __restrict__ esrc, float* __restrict__ out) {
    const int e = blockIdx.x * 256 + threadIdx.x;
    const int b = blockIdx.y;
    if (e >= NEDGE) return;
    const int s = esrc[e];
    const float4* wp = (const float4*)(w3 + (size_t)e * KCH);
    const float4* hp = (const float4*)(h + (size_t)b * CCH + (size_t)s * KCH);
    float4 w0 = wp[0], w1v = wp[1], w2 = wp[2], w3v = wp[3];
    float4 h0 = hp[0], h1 = hp[1], h2 = hp[2], h3 = hp[3];
    float acc = b3[e] + x[(size_t)b * NEDGE + e];
    acc += w0.x * h0.x + w0.y * h0.y + w0.z * h0.z + w0.w * h0.w;
    acc += w1v.x * h1.x + w1v.y * h1.y + w1v.z * h1.z + w1v.w * h1.w;
    acc += w2.x * h2.x + w2.y * h2.y + w2.z * h2.z + w2.w * h2.w;
    acc += w3v.x * h3.x + w3v.y * h3.y + w3v.z * h3.z + w3v.w * h3.w;
    out[(size_t)b * NEDGE + e] = acc;
}

// ---------------- host launcher ----------------

extern "C" void kernel_launch(void* const* d_in, const int* in_sizes, int n_in,
                              void* d_out, int out_size, void* d_ws, size_t ws_size,
                              hipStream_t stream) {
    (void)in_sizes; (void)n_in; (void)out_size;

    const float* x       = (const float*)d_in[0];
    const float* w1_vals = (const float*)d_in[1];
    const float* b1      = (const float*)d_in[2];
    const float* w3_vals = (const float*)d_in[3];
    const float* b3      = (const float*)d_in[4];
    const float* gamma   = (const float*)d_in[5];
    const float* beta    = (const float*)d_in[6];
    const int*   w1_dst  = (const int*)d_in[8];
    const int*   w3_src  = (const int*)d_in[9];
    float* out = (float*)d_out;

    // workspace carve-up (256B aligned slices)
    char* base = (char*)d_ws;
    size_t off = 0;
    auto alloc = [&](size_t bytes) -> void* {
        void* p = base + off;
        off = (off + bytes + 255) & ~(size_t)255;
        return p;
    };
    float* h        = (float*)alloc(sizeof(float) * (size_t)BATCH * CCH);
    int*   edge_dst = (int*)alloc(sizeof(int) * NEDGE);
    int*   edge_src = (int*)alloc(sizeof(int) * NEDGE);
    int*   csr      = (int*)alloc(sizeof(int) * NEDGE);
    int*   rowptr   = (int*)alloc(sizeof(int) * (NNODE + 1));
    int*   deg      = (int*)alloc(sizeof(int) * NNODE);
    int*   cnt      = (int*)alloc(sizeof(int) * NNODE);
    float* xT       = (float*)(base + off);
    const bool use_xt = ws_size >= off + sizeof(float) * (size_t)NEDGE * BATCH;

    const int gE = (NEDGE + 255) / 256;
    const int gN = (NNODE + 255) / 256;

    k_zero_i32<<<gN, 256, 0, stream>>>(deg, NNODE);
    k_zero_i32<<<gN, 256, 0, stream>>>(cnt, NNODE);
    if (use_xt)
        k_transpose<<<dim3((NEDGE + 31) / 32, BATCH / 32), dim3(32, 8), 0, stream>>>(x, xT);
    k_prep<<<gE, 256, 0, stream>>>(w1_dst, w3_src, edge_dst, edge_src, deg);
    k_scan<<<1, 32, 0, stream>>>(deg, rowptr);
    k_fill<<<gE, 256, 0, stream>>>(edge_dst, rowptr, cnt, csr);
    k_sort<<<gN, 256, 0, stream>>>(rowptr, csr);
    if (use_xt)
        k_lin1_norm<true><<<dim3(NNODE, BATCH / 16), 32, 0, stream>>>(
            xT, w1_vals, b1, gamma, beta, rowptr, csr, h);
    else
        k_lin1_norm<false><<<dim3(NNODE, BATCH / 16), 32, 0, stream>>>(
            x, w1_vals, b1, gamma, beta, rowptr, csr, h);
    k_lin3<<<dim3(gE, BATCH), 256, 0, stream>>>(x, h, w3_vals, b3, edge_src, out);
}